// Model_29188597744244
// MI455X (gfx1250) — compile-verified
//
#include <hip/hip_runtime.h>
#include <hip/hip_bf16.h>
#include <math.h>

typedef __attribute__((ext_vector_type(2))) float v2f;
typedef __attribute__((ext_vector_type(8))) float v8f;

#define F_IN 128
#define NHID 64
#define NCLS 16

// ---------------------------------------------------------------------------
// Degree / normalization
// ---------------------------------------------------------------------------
__global__ void k_init_deg(float* __restrict__ deg, int N) {
    int i = blockIdx.x * blockDim.x + threadIdx.x;
    if (i < N) deg[i] = 1.0f;  // self loop
}

__global__ void k_accum_deg(const long long* __restrict__ col,
                            float* __restrict__ deg, int E, int N) {
    int e = blockIdx.x * blockDim.x + threadIdx.x;
    if (e < E) {
        int c = (int)col[e];
        if ((unsigned)c < (unsigned)N) atomicAdd(&deg[c], 1.0f);
    }
}

__global__ void k_rsqrt_deg(float* __restrict__ d, int N) {
    int i = blockIdx.x * blockDim.x + threadIdx.x;
    if (i < N) d[i] = rsqrtf(d[i]);  // deg >= 1 always (self loops)
}

// ---------------------------------------------------------------------------
// Dense GEMM  T[N,OUT] = (RELU? max(X,0):X)[N,K] @ W[K,OUT]
// using V_WMMA_F32_16X16X4_F32. One wave owns a 16x16 output tile.
// A(16x4 f32): lanes 0-15 hold M=0..15,(K,K+1); lanes 16-31 hold (K+2,K+3).
// B(4x16 f32): v0: lanes0-15=rowK, lanes16-31=rowK+2; v1: rowK+1 / rowK+3.
// C/D(16x16):  VGPR r -> M=r (lanes 0-15), M=r+8 (lanes 16-31), N=lane&15.
// ---------------------------------------------------------------------------
template <int K, int OUT, bool RELU>
__global__ void k_gemm(const float* __restrict__ X, const float* __restrict__ W,
                       float* __restrict__ T, int N) {
    constexpr int CT = OUT / 16;          // column tiles per row tile
    const int wave = threadIdx.x >> 5;    // 4 waves / block (wave32)
    const int lane = threadIdx.x & 31;
    const int rowTile = blockIdx.x * (4 / CT) + wave / CT;
    const int colBase = (wave % CT) * 16;
    const int rowBase = rowTile * 16;
    if (rowBase >= N) return;             // N % 16 == 0, so full tiles only

    const int lhalf = lane >> 4;          // 0: K,K+1   1: K+2,K+3
    const int lm    = lane & 15;

    const float* ap = X + (size_t)(rowBase + lm) * K + 2 * lhalf;
    const float* bp = W + (size_t)(2 * lhalf) * OUT + colBase + lm;

    v8f acc = {};
#pragma unroll
    for (int k = 0; k < K; k += 4) {
        v2f a;
        a.x = ap[0];
        a.y = ap[1];
        if (RELU) { a.x = fmaxf(a.x, 0.0f); a.y = fmaxf(a.y, 0.0f); }
        v2f b;
        b.x = bp[0];
        b.y = bp[OUT];
        acc = __builtin_amdgcn_wmma_f32_16x16x4_f32(
            /*neg_a=*/false, a, /*neg_b=*/false, b,
            /*c_mod=*/(short)0, acc, /*reuse_a=*/false, /*reuse_b=*/false);
        ap += 4;
        bp += 4 * OUT;
    }

    float* op = T + (size_t)(rowBase + 8 * lhalf) * OUT + colBase + lm;
#pragma unroll
    for (int r = 0; r < 8; ++r) op[(size_t)r * OUT] = acc[r];
}

// ---------------------------------------------------------------------------
// out[i,f] = T[i,f] * dis[i]^2 (self-loop term) + b[f]
// ---------------------------------------------------------------------------
template <int OUT>
__global__ void k_init_out(const float* __restrict__ T,
                           const float* __restrict__ dis,
                           const float* __restrict__ b,
                           float* __restrict__ Out, int N) {
    int idx = blockIdx.x * blockDim.x + threadIdx.x;
    if (idx < N * OUT) {
        int i = idx / OUT;
        int f = idx % OUT;
        float d = dis[i];
        Out[idx] = T[idx] * d * d + b[f];
    }
}

// ---------------------------------------------------------------------------
// Edge scatter: Out[col,f] += T[row,f] * dis[row] * dis[col]
// 32 consecutive lanes handle consecutive f of one edge -> coalesced
// L2-resident gathers + coalesced global_atomic_add_f32.
// ---------------------------------------------------------------------------
template <int OUT>
__global__ void k_scatter(const float* __restrict__ T,
                          const float* __restrict__ dis,
                          const long long* __restrict__ row,
                          const long long* __restrict__ col,
                          float* __restrict__ Out, int E) {
    long long idx = (long long)blockIdx.x * blockDim.x + threadIdx.x;
    if (idx >= (long long)E * OUT) return;
    int e = (int)(idx / OUT);
    int f = (int)(idx & (OUT - 1));
    int r = (int)row[e];
    int c = (int)col[e];
    float w = dis[r] * dis[c];
    atomicAdd(&Out[(size_t)c * OUT + f], T[(size_t)r * OUT + f] * w);
}

// ---------------------------------------------------------------------------
// Row-wise log_softmax over NCLS=16, in place.
// ---------------------------------------------------------------------------
__global__ void k_logsoftmax(float* __restrict__ out, int N) {
    int i = blockIdx.x * blockDim.x + threadIdx.x;
    if (i >= N) return;
    float* p = out + (size_t)i * NCLS;
    float v[NCLS];
    float mx = -INFINITY;
#pragma unroll
    for (int f = 0; f < NCLS; ++f) { v[f] = p[f]; mx = fmaxf(mx, v[f]); }
    float s = 0.0f;
#pragma unroll
    for (int f = 0; f < NCLS; ++f) s += expf(v[f] - mx);
    float ls = logf(s);
#pragma unroll
    for (int f = 0; f < NCLS; ++f) p[f] = v[f] - mx - ls;
}

// ---------------------------------------------------------------------------
extern "C" void kernel_launch(void* const* d_in, const int* in_sizes, int n_in,
                              void* d_out, int out_size, void* d_ws, size_t ws_size,
                              hipStream_t stream) {
    const float*     x   = (const float*)d_in[0];
    const long long* ei  = (const long long*)d_in[1];  // int64 edge_index [2,E]
    const float*     W1  = (const float*)d_in[2];
    const float*     b1  = (const float*)d_in[3];
    const float*     W2  = (const float*)d_in[4];
    const float*     b2  = (const float*)d_in[5];
    const float*     W3  = (const float*)d_in[6];
    const float*     b3  = (const float*)d_in[7];
    float* out = (float*)d_out;

    const int N = in_sizes[0] / F_IN;   // 50000
    const int E = in_sizes[1] / 2;      // 800000
    const long long* erow = ei;         // source j
    const long long* ecol = ei + E;     // target i

    // workspace layout
    char* ws = (char*)d_ws;
    size_t off = 0;
    float* dis = (float*)(ws + off);            off += ((size_t)N * 4 + 255) & ~(size_t)255;
    float* bufA = (float*)(ws + off);           off += (size_t)N * NHID * 4;   // t (pre-agg)
    float* bufB = (float*)(ws + off);           off += (size_t)N * NHID * 4;   // h (post-agg)
    float* bufC = (float*)(ws + off);           off += (size_t)N * NCLS * 4;   // t3

    const int TB = 256;
    const int nb_N   = (N + TB - 1) / TB;
    const int nb_E   = (E + TB - 1) / TB;
    const int nb_N64 = (N * NHID + TB - 1) / TB;
    const int nb_N16 = (N * NCLS + TB - 1) / TB;
    const int nb_E64 = (int)(((long long)E * NHID + TB - 1) / TB);
    const int nb_E16 = (int)(((long long)E * NCLS + TB - 1) / TB);
    const int gemm64_blocks = N / 16;                 // 3125 (N % 16 == 0)
    const int gemm16_blocks = (N / 16 + 3) / 4;       // 4 row tiles per block

    // normalization: deg -> dis
    k_init_deg<<<nb_N, TB, 0, stream>>>(dis, N);
    k_accum_deg<<<nb_E, TB, 0, stream>>>(ecol, dis, E, N);
    k_rsqrt_deg<<<nb_N, TB, 0, stream>>>(dis, N);

    // layer 1: t = x @ W1 ; h1 = D^-1/2 A D^-1/2 t + b1  (relu folded into L2 load)
    k_gemm<F_IN, NHID, false><<<gemm64_blocks, 128, 0, stream>>>(x, W1, bufA, N);
    k_init_out<NHID><<<nb_N64, TB, 0, stream>>>(bufA, dis, b1, bufB, N);
    k_scatter<NHID><<<nb_E64, TB, 0, stream>>>(bufA, dis, erow, ecol, bufB, E);

    // layer 2: t = relu(h1) @ W2 ; h2 = agg(t) + b2
    k_gemm<NHID, NHID, true><<<gemm64_blocks, 128, 0, stream>>>(bufB, W2, bufA, N);
    k_init_out<NHID><<<nb_N64, TB, 0, stream>>>(bufA, dis, b2, bufB, N);
    k_scatter<NHID><<<nb_E64, TB, 0, stream>>>(bufA, dis, erow, ecol, bufB, E);

    // layer 3: t3 = h2 @ W3 ; logits = agg(t3) + b3
    k_gemm<NHID, NCLS, false><<<gemm16_blocks, 128, 0, stream>>>(bufB, W3, bufC, N);
    k_init_out<NCLS><<<nb_N16, TB, 0, stream>>>(bufC, dis, b3, out, N);
    k_scatter<NCLS><<<nb_E16, TB, 0, stream>>>(bufC, dis, erow, ecol, out, E);

    // log_softmax rows of 16
    k_logsoftmax<<<nb_N, TB, 0, stream>>>(out, N);

    (void)n_in; (void)out_size; (void)ws_size;
}